// ConvLayer_71373766525466
// MI455X (gfx1250) — compile-verified
//
#include <hip/hip_runtime.h>
#include <stdint.h>

typedef uint32_t u32;
typedef uint64_t u64;
typedef __attribute__((ext_vector_type(4))) u32 su4;
typedef __attribute__((ext_vector_type(8))) u32 su8;

#define TS   64               // output tile edge
#define IW   70               // input tile edge (TS + 6 halo)
#define LDSN (IW * IW)        // 4900 floats = 19.6 KB

__global__ __launch_bounds__(256)
void dwconv7_mish_kernel(const float* __restrict__ xin,
                         const float* __restrict__ kern,
                         float* __restrict__ yout)
{
    __shared__ float tile[LDSN];

    const u32 tid   = threadIdx.x;
    const u32 blk   = blockIdx.x;
    const u32 t     = blk & 15u;        // 4x4 tiles per 256x256 plane
    const u32 plane = blk >> 4;         // b*64 + c, 0..1023
    const int x0 = (int)(t & 3u) * TS;
    const int y0 = (int)(t >> 2) * TS;

    // Valid (in-bounds) sub-rectangle of the 70x70 halo window.
    const int xs = max(x0 - 3, 0);
    const int ys = max(y0 - 3, 0);
    const int xe = min(x0 + TS + 3, 256);
    const int ye = min(y0 + TS + 3, 256);
    const int w  = xe - xs;             // 67..70
    const int h  = ye - ys;             // 67..70
    const int lox = xs - (x0 - 3);      // where valid region lands in LDS tile
    const int loy = ys - (y0 - 3);

    // ---------------- wave 0: one TDM op DMAs the tile into LDS ----------------
    // Iterate mode: each iteration moves one w-float row; LDS advances by a
    // fixed stride of IW floats so border-clipped tiles keep the same layout.
    if (tid < 32u) {
        const u64 gaddr = (u64)(uintptr_t)xin +
                          ((((u64)plane << 16) + (u64)(ys * 256 + xs)) << 2);
        const u32 lds0  = (u32)(uintptr_t)(&tile[loy * IW + lox]); // low 32b = LDS byte offset

        su4 g0; su8 g1; su4 g2; su4 g3;
        // D# group 0: count=1 | lds_addr | global_addr[56:0] | type=2
        g0[0] = 1u;
        g0[1] = lds0;
        g0[2] = (u32)gaddr;
        g0[3] = ((u32)(gaddr >> 32) & 0x01FFFFFFu) | 0x80000000u;
        // D# group 1: data_size=4B (2<<16), iterate_enable (1<<19)
        g1[0] = (2u << 16) | (1u << 19);
        g1[1] = ((u32)w & 0xFFFFu) << 16;                       // tensor_dim0[15:0]
        g1[2] = ((u32)w >> 16) | (((u32)h & 0xFFFFu) << 16);    // td0 hi | tensor_dim1 lo
        g1[3] = ((u32)h >> 16) | ((u32)w << 16);                // td1 hi | tile_dim0 = w
        g1[4] = 1u;                                             // tile_dim1=1, tile_dim2=0
        g1[5] = 256u;                                           // tensor_dim0_stride lo
        g1[6] = 0u;                                             // td0_stride hi | td1_stride lo
        g1[7] = 0u;
        // D# group 2 (iterate mode): lds_inc=IW elems, global_inc=256 elems, count=h-1
        g2[0] = 0u;
        g2[1] = (u32)IW;
        g2[2] = 256u;
        g2[3] = ((u32)(h - 1)) << 16;
        g3[0] = 0u; g3[1] = 0u; g3[2] = 0u; g3[3] = 0u;
#if defined(__HIP_DEVICE_COMPILE__)
        asm volatile("tensor_load_to_lds %0, %1, %2, %3"
                     :: "s"(g0), "s"(g1), "s"(g2), "s"(g3) : "memory");
#endif
    }

    // Border tiles: zero ONLY the out-of-bounds halo cells (disjoint from the
    // region the TDM writes, so no ordering against the DMA is required).
    if ((x0 == 0) | (y0 == 0) | (x0 + TS == 256) | (y0 + TS == 256)) {
        for (u32 i = tid; i < (u32)LDSN; i += 256u) {
            const int ly = (int)(i / IW);
            const int lx = (int)(i - (u32)ly * (u32)IW);
            const int gy = y0 - 3 + ly;
            const int gx = x0 - 3 + lx;
            if ((unsigned)gy >= 256u || (unsigned)gx >= 256u) tile[i] = 0.0f;
        }
    }

    // Flipped kernel (true convolution) hoisted into SGPRs: wave-uniform, so
    // every conv FMA is v_fmac with a scalar operand (no VGPR cost).
    float kf[7][7];
#pragma unroll
    for (int a = 0; a < 7; ++a)
#pragma unroll
        for (int b = 0; b < 7; ++b)
            kf[a][b] = __int_as_float(__builtin_amdgcn_readfirstlane(
                           __float_as_int(kern[(6 - a) * 7 + (6 - b)])));

    if (tid < 32u) __builtin_amdgcn_s_wait_tensorcnt(0);
    __syncthreads();

    // ---------------- compute: 1 column x 16 rows per thread ----------------
    const u32 tx = tid & 63u;           // column within tile
    const u32 r0 = (tid >> 6) * 16u;    // first output row within tile

    float acc[16];
#pragma unroll
    for (int m = 0; m < 16; ++m) acc[m] = 0.0f;

    const float* lp = &tile[r0 * IW + tx];
#pragma unroll
    for (int ir = 0; ir < 22; ++ir) {   // sliding 22-row input window
        float v[7];
#pragma unroll
        for (int j = 0; j < 7; ++j) v[j] = lp[ir * IW + j];
#pragma unroll
        for (int m = 0; m < 16; ++m) {
            const int a = ir - m;       // vertical tap, constant after unroll
            if (a >= 0 && a < 7) {
                float s = acc[m];
#pragma unroll
                for (int j = 0; j < 7; ++j) s = fmaf(kf[a][j], v[j], s);
                acc[m] = s;
            }
        }
    }

    // mish(x) = x * ((1+e^x)^2 - 1) / ((1+e^x)^2 + 1); saturate for large x.
    float* op = yout + (((u64)plane << 16) +
                        (u64)(y0 + (int)r0) * 256u + (u64)(x0 + (int)tx));
#pragma unroll
    for (int m = 0; m < 16; ++m) {
        const float yv = acc[m];
        const float e  = __expf(yv);
        const float p  = 1.0f + e;
        const float q  = p * p;
        const float r  = (yv > 15.0f)
                           ? yv
                           : yv * (q - 1.0f) * __builtin_amdgcn_rcpf(q + 1.0f);
        op[(u64)m * 256u] = r;
    }
}

extern "C" void kernel_launch(void* const* d_in, const int* in_sizes, int n_in,
                              void* d_out, int out_size, void* d_ws, size_t ws_size,
                              hipStream_t stream) {
    const float* x = (const float*)d_in[0];   // [16,64,256,256] fp32
    const float* k = (const float*)d_in[1];   // [7,7] fp32
    float* out     = (float*)d_out;           // [16,64,256,256] fp32
    // 1024 planes * 16 tiles = 16384 blocks, 256 threads (8 wave32) each.
    dwconv7_mish_kernel<<<dim3(16384), dim3(256), 0, stream>>>(x, k, out);
}